// GIN_21251498181409
// MI455X (gfx1250) — compile-verified
//
#include <hip/hip_runtime.h>
#include <math.h>

#define BATCH   16
#define NN      2048
#define HD      128
#define NODES   (BATCH*NN)      // 32768
#define NCLS    21
#define SPLIT   4
#define JCHUNK  (NN/SPLIT)      // 512
#define SCAP    32              // per-split edge capacity (mean deg 8/split)
#define ECAP    (SPLIT*SCAP)    // 128 per target
#define BN_EPS  1e-5f

typedef __attribute__((ext_vector_type(2))) float v2f;
typedef __attribute__((ext_vector_type(8))) float v8f;

// ---------------------------------------------------------------------------
// 1) Deterministic CSC build: one thread scans a 512-row chunk of one column
//    of A (fully coalesced across lanes: consecutive lanes -> consecutive i).
// ---------------------------------------------------------------------------
__global__ __launch_bounds__(256)
void k_build(const float* __restrict__ A, int* __restrict__ deg,
             int* __restrict__ edges) {
  int gt = blockIdx.x * blockDim.x + threadIdx.x;   // 0 .. SPLIT*NODES-1
  int s  = gt / NODES;                              // split id (grid-major)
  int t  = gt - s * NODES;                          // target node = b*NN + i
  int b  = t >> 11;
  int i  = t & (NN - 1);
  const float* col = A + (size_t)b * NN * NN + i;
  int jbase = s * JCHUNK;
  int cnt = 0;
  int* elist = edges + (size_t)t * ECAP + s * SCAP;
  for (int j = 0; j < JCHUNK; j += 4) {
    float a0 = col[(size_t)(jbase + j + 0) * NN];
    float a1 = col[(size_t)(jbase + j + 1) * NN];
    float a2 = col[(size_t)(jbase + j + 2) * NN];
    float a3 = col[(size_t)(jbase + j + 3) * NN];
    if (a0 != 0.f && cnt < SCAP) elist[cnt++] = jbase + j + 0;
    if (a1 != 0.f && cnt < SCAP) elist[cnt++] = jbase + j + 1;
    if (a2 != 0.f && cnt < SCAP) elist[cnt++] = jbase + j + 2;
    if (a3 != 0.f && cnt < SCAP) elist[cnt++] = jbase + j + 3;
  }
  deg[t * SPLIT + s] = cnt;
}

// ---------------------------------------------------------------------------
// 2) Aggregation: z = (1+eps)*h + sum_{j in col(i)} h[j].  128 threads/node,
//    each gather is a 512B coalesced row read (h is L2-resident, 16MB << 192MB).
// ---------------------------------------------------------------------------
__global__ __launch_bounds__(256)
void k_agg(const float* __restrict__ hin, const int* __restrict__ deg,
           const int* __restrict__ edges, const float* __restrict__ epsp,
           float* __restrict__ zout) {
  int t = blockIdx.x * 2 + (threadIdx.x >> 7);
  int f = threadIdx.x & (HD - 1);
  int b = t >> 11;
  const float* hb = hin + (size_t)b * NN * HD + f;
  float acc = 0.f;
  const int* dg = deg + t * SPLIT;
  const int* el = edges + (size_t)t * ECAP;
  for (int s = 0; s < SPLIT; ++s) {
    int n = dg[s];
    const int* e = el + s * SCAP;
    for (int c = 0; c < n; ++c)
      acc += hb[(size_t)e[c] * HD];
  }
  float eps  = *epsp;
  float self = hin[(size_t)t * HD + f];
  zout[(size_t)t * HD + f] = (1.f + eps) * self + acc;
}

// ---------------------------------------------------------------------------
// 3) Pack 128x128 fp32 weights into the V_WMMA_F32_16X16X4_F32 B-fragment
//    layout: out[((kt*8+nt)*32+lane)*2+j] = W[(kt*4 + (lane>=16?2:0) + j)*128
//                                             + nt*16 + (lane&15)]
// ---------------------------------------------------------------------------
struct WPtrs { const float* w[9]; };

__global__ __launch_bounds__(256)
void k_pack(WPtrs wps, float* __restrict__ out) {
  int m = blockIdx.y;
  int o = blockIdx.x * blockDim.x + threadIdx.x;    // 0..16383
  int j    =  o        & 1;
  int lane = (o >> 1)  & 31;
  int nt   = (o >> 6)  & 7;
  int kt   =  o >> 9;
  int k = kt * 4 + ((lane >> 4) << 1) + j;
  int n = nt * 16 + (lane & 15);
  out[(size_t)m * HD * HD + o] = wps.w[m][k * HD + n];
}

// ---------------------------------------------------------------------------
// 4) GEMM: Y[32768x128] = epilogue(X[32768x128] @ W[128x128] + bias)
//    fp32 WMMA 16x16x4, K=128 -> 32 steps. Wave tile = 16(M) x 64(N):
//    4 independent accumulator chains. Block = 8 waves = 64 x 128.
//    mode 0: relu(x+b);  mode 1: relu -> BN affine -> relu.
// ---------------------------------------------------------------------------
__global__ __launch_bounds__(256)
void k_gemm(const float* __restrict__ X, const float* __restrict__ Wp,
            const float* __restrict__ bias, const float* __restrict__ gamma,
            const float* __restrict__ beta, const float* __restrict__ mean,
            const float* __restrict__ var, int mode, float* __restrict__ Y) {
  int lane = threadIdx.x & 31;
  int wid  = threadIdx.x >> 5;
  int mt = blockIdx.x * 64 + (wid & 3) * 16;
  int nb = (wid >> 2) << 6;                 // 0 or 64
  int row = mt + (lane & 15);
  int khalf = (lane >> 4) << 1;             // lanes 16-31 hold K=2,3
  const float* xrow = X + (size_t)row * HD + khalf;
  int nt0 = nb >> 4;                        // 0 or 4

  v8f acc[4] = {};
  for (int kt = 0; kt < 32; ++kt) {
    v2f a = *(const v2f*)(xrow + kt * 4);
    const float* wk = Wp + (((size_t)kt * 8 + nt0) * 32 + lane) * 2;
    v2f b0 = *(const v2f*)(wk);
    v2f b1 = *(const v2f*)(wk + 64);
    v2f b2 = *(const v2f*)(wk + 128);
    v2f b3 = *(const v2f*)(wk + 192);
    acc[0] = __builtin_amdgcn_wmma_f32_16x16x4_f32(false, a, false, b0, (short)0, acc[0], false, false);
    acc[1] = __builtin_amdgcn_wmma_f32_16x16x4_f32(false, a, false, b1, (short)0, acc[1], false, false);
    acc[2] = __builtin_amdgcn_wmma_f32_16x16x4_f32(false, a, false, b2, (short)0, acc[2], false, false);
    acc[3] = __builtin_amdgcn_wmma_f32_16x16x4_f32(false, a, false, b3, (short)0, acc[3], false, false);
  }

  // C layout: VGPR r -> M = mt + r + (lane>=16 ? 8 : 0), N = nb + j*16 + (lane&15)
  int mrow = mt + ((lane >> 4) << 3);
  for (int j = 0; j < 4; ++j) {
    int n = nb + j * 16 + (lane & 15);
    float bb = bias[n];
    float g = 0.f, bt = 0.f, mu = 0.f, rs = 0.f;
    if (mode) {
      g = gamma[n]; bt = beta[n]; mu = mean[n];
      rs = rsqrtf(var[n] + BN_EPS);
    }
    float* yp = Y + (size_t)mrow * HD + n;
    #pragma unroll
    for (int r = 0; r < 8; ++r) {
      float v = acc[j][r] + bb;
      v = fmaxf(v, 0.f);
      if (mode) { v = g * (v - mu) * rs + bt; v = fmaxf(v, 0.f); }
      yp[(size_t)r * HD] = v;
    }
  }
}

// ---------------------------------------------------------------------------
// 5) Head: logits = Y @ lin2_W + b  (N=21), fused log_softmax.
//    One wave per node; row staged in LDS; wave shuffles for max / sum-exp.
// ---------------------------------------------------------------------------
__global__ __launch_bounds__(256)
void k_head(const float* __restrict__ Yin, const float* __restrict__ W,
            const float* __restrict__ bias, float* __restrict__ lsm,
            float* __restrict__ logits) {
  __shared__ float rows[8][HD];
  int wid  = threadIdx.x >> 5;
  int lane = threadIdx.x & 31;
  int t = blockIdx.x * 8 + wid;
  for (int f = lane; f < HD; f += 32)
    rows[wid][f] = Yin[(size_t)t * HD + f];
  __syncthreads();

  float logit = -INFINITY;
  if (lane < NCLS) {
    float a = bias[lane];
    for (int f = 0; f < HD; ++f)
      a = fmaf(rows[wid][f], W[f * NCLS + lane], a);
    logit = a;
  }
  float mx = logit;
  for (int off = 16; off; off >>= 1)
    mx = fmaxf(mx, __shfl_xor(mx, off));
  float ex = (lane < NCLS) ? expf(logit - mx) : 0.f;
  float se = ex;
  for (int off = 16; off; off >>= 1)
    se += __shfl_xor(se, off);
  float lse = logf(se) + mx;
  if (lane < NCLS) {
    lsm   [(size_t)t * NCLS + lane] = logit - lse;
    logits[(size_t)t * NCLS + lane] = logit;
  }
}

// ---------------------------------------------------------------------------
extern "C" void kernel_launch(void* const* d_in, const int* in_sizes, int n_in,
                              void* d_out, int out_size, void* d_ws, size_t ws_size,
                              hipStream_t stream) {
  // --- identify inputs by size (robust to dict-flattening order) ---
  int idx_x = -1, idx_A = -1;
  for (int i = 0; i < n_in; ++i) {
    if (in_sizes[i] == NODES * HD)          idx_x = i;   // 4,194,304
    if (in_sizes[i] == BATCH * NN * NN)     idx_A = i;   // 67,108,864
  }
  int P[64]; int np = 0;
  for (int i = 0; i < n_in; ++i)
    if (i != idx_x && i != idx_A) P[np++] = i;
  // P: 4 conv groups of 9 tensors, then lin1_W, lin1_b, lin2_W, lin2_b

  struct Conv { const float *eps,*W1,*b1,*W2,*b2,*gamma,*beta,*mean,*var; } cv[4];
  for (int l = 0; l < 4; ++l) {
    const int* g = P + 9 * l;
    if (in_sizes[g[0]] == 1) {
      // insertion order: eps,W1,b1,W2,b2,gamma,beta,mean,var
      cv[l] = { (const float*)d_in[g[0]], (const float*)d_in[g[1]],
                (const float*)d_in[g[2]], (const float*)d_in[g[3]],
                (const float*)d_in[g[4]], (const float*)d_in[g[5]],
                (const float*)d_in[g[6]], (const float*)d_in[g[7]],
                (const float*)d_in[g[8]] };
    } else {
      // key-sorted order: W1,W2,b1,b2,beta,eps,gamma,mean,var
      cv[l].W1    = (const float*)d_in[g[0]];
      cv[l].W2    = (const float*)d_in[g[1]];
      cv[l].b1    = (const float*)d_in[g[2]];
      cv[l].b2    = (const float*)d_in[g[3]];
      cv[l].beta  = (const float*)d_in[g[4]];
      cv[l].eps   = (const float*)d_in[g[5]];
      cv[l].gamma = (const float*)d_in[g[6]];
      cv[l].mean  = (const float*)d_in[g[7]];
      cv[l].var   = (const float*)d_in[g[8]];
    }
  }
  const float* lin1W = (const float*)d_in[P[36]];
  const float* lin1b = (const float*)d_in[P[37]];
  const float* lin2W = (const float*)d_in[P[38]];
  const float* lin2b = (const float*)d_in[P[39]];
  const float* xf = (const float*)d_in[idx_x];
  const float* Af = (const float*)d_in[idx_A];

  // --- workspace carve-out (~65 MB) ---
  char* ws = (char*)d_ws;
  int*   deg   = (int*)ws;                                   // 512 KB
  int*   edges = (int*)(ws + (size_t)(NODES * SPLIT) * 4);   // 16 MB
  float* h     = (float*)(ws + (size_t)(NODES * SPLIT) * 4
                             + (size_t)NODES * ECAP * 4);
  float* zbuf  = h    + (size_t)NODES * HD;                  // 16 MB each
  float* ybuf  = zbuf + (size_t)NODES * HD;
  float* Wp    = ybuf + (size_t)NODES * HD;                  // 9 * 64 KB

  // --- build adjacency (reads A exactly once, coalesced) ---
  k_build<<<(SPLIT * NODES) / 256, 256, 0, stream>>>(Af, deg, edges);

  // --- pack all 9 GEMM weight matrices into WMMA fragment layout ---
  WPtrs wps;
  for (int l = 0; l < 4; ++l) { wps.w[2*l] = cv[l].W1; wps.w[2*l+1] = cv[l].W2; }
  wps.w[8] = lin1W;
  k_pack<<<dim3(HD * HD / 256, 9), 256, 0, stream>>>(wps, Wp);

  // --- 4 GIN conv layers ---
  const float* cur = xf;
  for (int l = 0; l < 4; ++l) {
    k_agg<<<NODES / 2, 256, 0, stream>>>(cur, deg, edges, cv[l].eps, zbuf);
    k_gemm<<<NODES / 64, 256, 0, stream>>>(zbuf, Wp + (size_t)(2*l) * HD * HD,
        cv[l].b1, nullptr, nullptr, nullptr, nullptr, 0, ybuf);
    k_gemm<<<NODES / 64, 256, 0, stream>>>(ybuf, Wp + (size_t)(2*l+1) * HD * HD,
        cv[l].b2, cv[l].gamma, cv[l].beta, cv[l].mean, cv[l].var, 1, h);
    cur = h;
  }

  // --- head: lin1 + relu (WMMA), then lin2 + log_softmax fused ---
  k_gemm<<<NODES / 64, 256, 0, stream>>>(h, Wp + (size_t)8 * HD * HD,
      lin1b, nullptr, nullptr, nullptr, nullptr, 0, ybuf);
  float* out_lsm = (float*)d_out;
  float* out_log = out_lsm + (size_t)NODES * NCLS;
  k_head<<<NODES / 8, 256, 0, stream>>>(ybuf, lin2W, lin2b, out_lsm, out_log);
}